// Cheby_74045236183289
// MI455X (gfx1250) — compile-verified
//
#include <hip/hip_runtime.h>
#include <hip/hip_bf16.h>

typedef __attribute__((ext_vector_type(2))) float v2f;
typedef __attribute__((ext_vector_type(8))) float v8f;

#define D 256           // feature dim (DIN == DOUT)
#define D4 64           // feature dim in float4
#define LDS_STRIDE 260  // 1040B row stride -> 4-bank rotation, conflict-free b64 reads

// ---------------------------------------------------------------------------
// Degree / normalization kernels
// ---------------------------------------------------------------------------
__global__ void zero_deg_kernel(int* __restrict__ deg, int n) {
    int i = blockIdx.x * blockDim.x + threadIdx.x;
    if (i < n) deg[i] = 0;
}

__global__ void count_deg_kernel(const int* __restrict__ row, int* __restrict__ deg, int nE) {
    int e = blockIdx.x * blockDim.x + threadIdx.x;
    if (e < nE) atomicAdd(&deg[row[e]], 1);
}

__global__ void calc_dis_kernel(const int* __restrict__ deg, float* __restrict__ dis, int n) {
    int i = blockIdx.x * blockDim.x + threadIdx.x;
    if (i < n) {
        // self-loop appended -> degree is count + 1, always >= 1
        dis[i] = rsqrtf((float)(deg[i] + 1));
    }
}

// ---------------------------------------------------------------------------
// Pack W[k] (256x256, row-major K x N) into per-lane WMMA fragment order:
//   pack4[((k*8 + wavecol)*64 + s)*32 + lane] =
//     { Wk[kA][n0], Wk[kA+1][n0], Wk[kA][n0+16], Wk[kA+1][n0+16] }
//   with kA = 4s + 2*(lane>>4), n0 = wavecol*32 + (lane&15)
// One b128 per K-step per wave then feeds two 16x16x4 WMMAs.
// ---------------------------------------------------------------------------
__global__ __launch_bounds__(256) void pack_w_kernel(const float* __restrict__ W,
                                                     float4* __restrict__ pack4) {
    int t = blockIdx.x * 256 + threadIdx.x;   // 0..65535
    int l   = t & 31;
    int s   = (t >> 5) & 63;
    int wc  = (t >> 11) & 7;
    int k   = t >> 14;
    int m   = l & 15;
    int h   = l >> 4;
    int kA  = 4 * s + 2 * h;
    int n0  = wc * 32 + m;
    const float* Wk = W + (size_t)k * D * D;
    float4 o;
    o.x = Wk[(size_t)kA * D + n0];
    o.y = Wk[(size_t)(kA + 1) * D + n0];
    o.z = Wk[(size_t)kA * D + n0 + 16];
    o.w = Wk[(size_t)(kA + 1) * D + n0 + 16];
    pack4[t] = o;
}

// ---------------------------------------------------------------------------
// SpMM init pass: dst = scale*(1 - dis[i]^2)*src[i,:] - pscale*prev[i,:]
// ---------------------------------------------------------------------------
__global__ void cheb_init_kernel(const float* __restrict__ src,
                                 const float* __restrict__ prev,
                                 const float* __restrict__ dis,
                                 float* __restrict__ dst,
                                 float scale, float pscale, int nN) {
    int gid = blockIdx.x * blockDim.x + threadIdx.x;
    int total = nN * D4;
    if (gid >= total) return;
    int i  = gid >> 6;
    int f4 = gid & 63;
    float d = dis[i];
    float w = scale * (1.0f - d * d);
    size_t off = (size_t)i * D4 + f4;
    float4 v  = ((const float4*)src)[off];
    float4 pv = ((const float4*)prev)[off];
    float4 o;
    o.x = w * v.x - pscale * pv.x;
    o.y = w * v.y - pscale * pv.y;
    o.z = w * v.z - pscale * pv.z;
    o.w = w * v.w - pscale * pv.w;
    ((float4*)dst)[off] = o;
}

// ---------------------------------------------------------------------------
// SpMM edge pass: one wave32 per edge, 8 feats/lane, L2 float atomics.
// ---------------------------------------------------------------------------
__global__ __launch_bounds__(256) void cheb_edges_kernel(
    const float* __restrict__ src, const float* __restrict__ dis,
    const int* __restrict__ row, const int* __restrict__ col,
    float* __restrict__ dst, float scale, int nE) {
    int wid  = threadIdx.x >> 5;
    int lane = threadIdx.x & 31;
    int e = blockIdx.x * 8 + wid;
    if (e >= nE) return;
    int r = row[e];
    int c = col[e];
    float w = -dis[r] * dis[c];
    if (r == c) w += 1.0f;
    w *= scale;
    const float4* s = (const float4*)(src + (size_t)c * D);
    float* dp = dst + (size_t)r * D;
#pragma unroll
    for (int h = 0; h < 2; ++h) {
        int f4 = lane + h * 32;
        float4 v = s[f4];
        atomicAdd(dp + f4 * 4 + 0, w * v.x);
        atomicAdd(dp + f4 * 4 + 1, w * v.y);
        atomicAdd(dp + f4 * 4 + 2, w * v.z);
        atomicAdd(dp + f4 * 4 + 3, w * v.w);
    }
}

// ---------------------------------------------------------------------------
// Fused 4-term GEMM via V_WMMA_F32_16X16X4_F32.
// Block = 256 threads (8 waves) owns 32 rows x 256 cols; wave = 2x2 tiles.
// Inner K-step: 2 LDS b64 (A frags) + 1 global b128 (packed B) -> 4 WMMAs.
// Tail block (16 rows) takes the FULL=false path; branches are block-uniform
// so EXEC is all-ones for every WMMA.
// ---------------------------------------------------------------------------
template <bool FULL>
__device__ __forceinline__ void gemm_body(
    float (*As)[LDS_STRIDE],
    const float* __restrict__ T0, const float* __restrict__ T1,
    const float* __restrict__ T2, const float* __restrict__ T3,
    const float4* __restrict__ pack4, const float* __restrict__ bias,
    float* __restrict__ out, int rowBase, int tid, int wid, int m, int h) {
    const int lane = tid & 31;
    v8f acc00 = {}, acc01 = {}, acc10 = {}, acc11 = {};
    const float* Ts[4] = {T0, T1, T2, T3};

#pragma unroll
    for (int k = 0; k < 4; ++k) {
        __syncthreads();
        const float* Tk = Ts[k];
        const int nload = FULL ? 8 : 4;   // 32 (or 16) rows x 256 cols
#pragma unroll
        for (int i = 0; i < nload; ++i) {
            int idx = tid + i * 256;
            int r   = idx >> 6;
            int c4  = idx & 63;
            float4 v = *(const float4*)(Tk + ((size_t)(rowBase + r)) * D + c4 * 4);
            *(float4*)(&As[r][c4 * 4]) = v;
        }
        __syncthreads();

        const float4* p4 = pack4 + ((size_t)(k * 8 + wid) * 64 * 32) + lane;
#pragma unroll 4
        for (int s = 0; s < 64; ++s) {
            const int kA = 4 * s + 2 * h;
            v2f a0;
            a0.x = As[m][kA];
            a0.y = As[m][kA + 1];
            float4 bb = p4[(size_t)s * 32];      // coalesced b128: both B frags
            v2f b0; b0.x = bb.x; b0.y = bb.y;
            v2f b1; b1.x = bb.z; b1.y = bb.w;
            acc00 = __builtin_amdgcn_wmma_f32_16x16x4_f32(
                false, a0, false, b0, (short)0, acc00, false, false);
            acc01 = __builtin_amdgcn_wmma_f32_16x16x4_f32(
                false, a0, false, b1, (short)0, acc01, false, false);
            if (FULL) {
                v2f a1;
                a1.x = As[16 + m][kA];
                a1.y = As[16 + m][kA + 1];
                acc10 = __builtin_amdgcn_wmma_f32_16x16x4_f32(
                    false, a1, false, b0, (short)0, acc10, false, false);
                acc11 = __builtin_amdgcn_wmma_f32_16x16x4_f32(
                    false, a1, false, b1, (short)0, acc11, false, false);
            }
        }
    }

    // C/D layout: VGPR j -> M = j + 8*h, N = m
    const int colBase = wid * 32;
    const float bias0 = bias[colBase + m];
    const float bias1 = bias[colBase + 16 + m];
#pragma unroll
    for (int j = 0; j < 8; ++j) {
        int rr = rowBase + j + 8 * h;
        out[(size_t)rr * D + colBase + m]      = acc00[j] + bias0;
        out[(size_t)rr * D + colBase + 16 + m] = acc01[j] + bias1;
        if (FULL) {
            int rr1 = rr + 16;
            out[(size_t)rr1 * D + colBase + m]      = acc10[j] + bias0;
            out[(size_t)rr1 * D + colBase + 16 + m] = acc11[j] + bias1;
        }
    }
}

__global__ __launch_bounds__(256) void cheb_gemm_kernel(
    const float* __restrict__ T0, const float* __restrict__ T1,
    const float* __restrict__ T2, const float* __restrict__ T3,
    const float4* __restrict__ pack4, const float* __restrict__ bias,
    float* __restrict__ out, int nN) {
    __shared__ float As[32][LDS_STRIDE];
    const int tid = threadIdx.x;
    const int wid = tid >> 5;
    const int m   = tid & 15;
    const int h   = (tid >> 4) & 1;
    const int rowBase = blockIdx.x * 32;
    if (rowBase + 32 <= nN) {
        gemm_body<true>(As, T0, T1, T2, T3, pack4, bias, out, rowBase, tid, wid, m, h);
    } else {
        gemm_body<false>(As, T0, T1, T2, T3, pack4, bias, out, rowBase, tid, wid, m, h);
    }
}

// ---------------------------------------------------------------------------
// Launcher
// ---------------------------------------------------------------------------
extern "C" void kernel_launch(void* const* d_in, const int* in_sizes, int n_in,
                              void* d_out, int out_size, void* d_ws, size_t ws_size,
                              hipStream_t stream) {
    const float* x    = (const float*)d_in[0];   // [N, 256]
    const int*   ei   = (const int*)d_in[1];     // [2, E]
    const float* W    = (const float*)d_in[2];   // [4, 256, 256]
    const float* bias = (const float*)d_in[3];   // [256]
    float* out = (float*)d_out;

    const int N = in_sizes[0] / D;
    const int E = in_sizes[1] / 2;
    const int* row = ei;
    const int* col = ei + E;

    // workspace layout
    char* ws = (char*)d_ws;
    size_t termBytes = (size_t)N * D * sizeof(float);     // 51.2 MB each
    int*    deg   = (int*)ws;                              // N ints
    float*  dis   = (float*)(ws + (1u << 18));             // 256 KB offset
    float4* pack4 = (float4*)(ws + (1u << 19));            // 512 KB offset, 1 MB
    float*  T1    = (float*)(ws + (3u << 19));             // 1.5 MB offset
    float*  T2    = (float*)((char*)T1 + termBytes);
    float*  T3    = (float*)((char*)T2 + termBytes);

    const int bs = 256;
    // 0) pack W into WMMA fragment order (one-shot, 1 MB)
    pack_w_kernel<<<256, bs, 0, stream>>>(W, pack4);

    // 1) degrees + D^{-1/2}
    zero_deg_kernel<<<(N + bs - 1) / bs, bs, 0, stream>>>(deg, N);
    count_deg_kernel<<<(E + bs - 1) / bs, bs, 0, stream>>>(row, deg, E);
    calc_dis_kernel<<<(N + bs - 1) / bs, bs, 0, stream>>>(deg, dis, N);

    const int initGrid = (N * D4 + bs - 1) / bs;
    const int edgeGrid = (E + 7) / 8;

    // 2) T1 = L x
    cheb_init_kernel<<<initGrid, bs, 0, stream>>>(x, x, dis, T1, 1.0f, 0.0f, N);
    cheb_edges_kernel<<<edgeGrid, bs, 0, stream>>>(x, dis, row, col, T1, 1.0f, E);
    // 3) T2 = 2 L T1 - x
    cheb_init_kernel<<<initGrid, bs, 0, stream>>>(T1, x, dis, T2, 2.0f, 1.0f, N);
    cheb_edges_kernel<<<edgeGrid, bs, 0, stream>>>(T1, dis, row, col, T2, 2.0f, E);
    // 4) T3 = 2 L T2 - T1
    cheb_init_kernel<<<initGrid, bs, 0, stream>>>(T2, T1, dis, T3, 2.0f, 1.0f, N);
    cheb_edges_kernel<<<edgeGrid, bs, 0, stream>>>(T2, dis, row, col, T3, 2.0f, E);

    // 5) out = x@W0 + T1@W1 + T2@W2 + T3@W3 + bias  (fp32 WMMA)
    cheb_gemm_kernel<<<(N + 31) / 32, bs, 0, stream>>>(x, T1, T2, T3, pack4, bias, out, N);
}